// Embedding_23141283791210
// MI455X (gfx1250) — compile-verified
//
#include <hip/hip_runtime.h>
#include <stdint.h>

// Embedding gather: out[t, :] = table[ids[t], :]
//   ids:   [8192]  int32
//   table: [50257, 768] fp32
//   out:   [8192, 768]  fp32
//
// CDNA5 Tensor Data Mover, GATHER MODE:
//   - one single-wave workgroup per group of 16 tokens
//   - TENSOR_LOAD_TO_LDS (gather_mode=1, sixteen 16-bit row indices in
//     D# groups 2/3) DMAs 16 table rows (16 x 3072 B) into LDS
//   - TENSOR_STORE_FROM_LDS (normal 2D mode) streams the tile to output
//   - s_wait_tensorcnt orders load -> store; waves do only SGPR work
//   - partial final group is re-based to num_tokens-16 (overlap rows are
//     rewritten with identical data), so every block is straight-line code
//     (requires num_tokens >= 16; harness uses 8192)

#define VOCAB 50257
#define EMBED 768
#define ROWS_PER_TDM 16

typedef __attribute__((ext_vector_type(4))) unsigned int v4u;
typedef __attribute__((ext_vector_type(8))) int         v8i;
typedef __attribute__((ext_vector_type(4))) int         v4i;

#if defined(__HIP_DEVICE_COMPILE__) && defined(__gfx1250__) && \
    __has_builtin(__builtin_amdgcn_tensor_load_to_lds) &&      \
    __has_builtin(__builtin_amdgcn_tensor_store_from_lds) &&   \
    __has_builtin(__builtin_amdgcn_s_wait_tensorcnt)
#define USE_TDM 1
#else
#define USE_TDM 0
#endif

// therock-10.0 headers ship the 6-arg TDM builtin; ROCm 7.2 has the 5-arg form.
#if __has_include(<hip/amd_detail/amd_gfx1250_TDM.h>)
#define TDM_ARITY 6
#else
#define TDM_ARITY 5
#endif

__global__ __launch_bounds__(32) void embed_gather_kernel(
    const int* __restrict__ ids,
    const float* __restrict__ table,
    float* __restrict__ out,
    int num_tokens)
{
    __shared__ float tile[ROWS_PER_TDM * EMBED];  // 48 KB LDS staging tile

    // Re-base a partial final group onto the last full 16-token window.
    int base = blockIdx.x * ROWS_PER_TDM;
    if (base + ROWS_PER_TDM > num_tokens) base = num_tokens - ROWS_PER_TDM;

#if USE_TDM
    // ---- sixteen row ids via four uniform s_load_b128, straight-line pack ----
    const int4* idv = (const int4*)(ids + base);
    const int4 a = idv[0], b = idv[1], c = idv[2], d = idv[3];
    unsigned idx_pack[8];
    idx_pack[0] = ((unsigned)a.x & 0xFFFFu) | ((unsigned)a.y << 16);
    idx_pack[1] = ((unsigned)a.z & 0xFFFFu) | ((unsigned)a.w << 16);
    idx_pack[2] = ((unsigned)b.x & 0xFFFFu) | ((unsigned)b.y << 16);
    idx_pack[3] = ((unsigned)b.z & 0xFFFFu) | ((unsigned)b.w << 16);
    idx_pack[4] = ((unsigned)c.x & 0xFFFFu) | ((unsigned)c.y << 16);
    idx_pack[5] = ((unsigned)c.z & 0xFFFFu) | ((unsigned)c.w << 16);
    idx_pack[6] = ((unsigned)d.x & 0xFFFFu) | ((unsigned)d.y << 16);
    idx_pack[7] = ((unsigned)d.z & 0xFFFFu) | ((unsigned)d.w << 16);

    const unsigned lds_base = (unsigned)(uintptr_t)(void*)&tile[0];

    // ================= D# for gather-load (table -> LDS) =================
    const unsigned long long gaddr = (unsigned long long)(uintptr_t)table;
    v4u g0l;
    g0l.x = 1u                 // count = 1 (valid descriptor)
          | (0u << 30)         // gather_index_size = 0 (16-bit indices)
          | (1u << 31);        // gather_mode = 1
    g0l.y = lds_base;                                        // lds_addr
    g0l.z = (unsigned)(gaddr & 0xFFFFFFFFu);                 // global_addr[31:0]
    g0l.w = (unsigned)((gaddr >> 32) & 0x1FFFFFFu)           // global_addr[56:32]
          | (2u << 30);                                      // type = 2 (image)

    // Group 1 layout: [17:16] data_size=2 (4B) | [79:48] tensor_dim0
    //  | [111:80] tensor_dim1 | [127:112] tile_dim0 | [143:128] tile_dim1
    //  | [207:160] tensor_dim0_stride
    v8i g1l;
    g1l[0] = (int)(2u << 16);                  // data_size = 4 B
    g1l[1] = (int)((unsigned)EMBED << 16);     // tensor_dim0 lo16 = 768
    g1l[2] = (int)((unsigned)VOCAB << 16);     // dim0 hi=0 | tensor_dim1 lo16
    g1l[3] = (int)((unsigned)EMBED << 16);     // dim1 hi=0 | tile_dim0 = 768
    g1l[4] = ROWS_PER_TDM;                     // tile_dim1 = 16 indices
    g1l[5] = EMBED;                            // tensor_dim0_stride lo32
    g1l[6] = 0;
    g1l[7] = 0;

    v4i g2l, g3l;                              // gather row indices
    g2l[0] = (int)idx_pack[0]; g2l[1] = (int)idx_pack[1];
    g2l[2] = (int)idx_pack[2]; g2l[3] = (int)idx_pack[3];
    g3l[0] = (int)idx_pack[4]; g3l[1] = (int)idx_pack[5];
    g3l[2] = (int)idx_pack[6]; g3l[3] = (int)idx_pack[7];

    // ================= D# for 2D store (LDS -> out) =================
    const unsigned long long oaddr =
        (unsigned long long)(uintptr_t)out +
        (unsigned long long)base * (unsigned long long)(EMBED * 4);
    v4u g0s;
    g0s.x = 1u;                                              // count=1, normal mode
    g0s.y = lds_base;
    g0s.z = (unsigned)(oaddr & 0xFFFFFFFFu);
    g0s.w = (unsigned)((oaddr >> 32) & 0x1FFFFFFu) | (2u << 30);

    v8i g1s = g1l;
    // Output tensor: tensor_dim1 = num_tokens; tile_dim1 = 16 rows.
    g1s[2] = (int)(((unsigned)num_tokens & 0xFFFFu) << 16);
    g1s[3] = (int)(((unsigned)num_tokens >> 16) & 0xFFFFu)
           | (int)((unsigned)EMBED << 16);

    v4i zero4 = {0, 0, 0, 0};
#if TDM_ARITY == 6
    v8i zero8 = {0, 0, 0, 0, 0, 0, 0, 0};
    __builtin_amdgcn_tensor_load_to_lds(g0l, g1l, g2l, g3l, zero8, 0);
    __builtin_amdgcn_s_wait_tensorcnt(0);
    __builtin_amdgcn_tensor_store_from_lds(g0s, g1s, zero4, zero4, zero8, 0);
#else
    __builtin_amdgcn_tensor_load_to_lds(g0l, g1l, g2l, g3l, 0);
    __builtin_amdgcn_s_wait_tensorcnt(0);
    __builtin_amdgcn_tensor_store_from_lds(g0s, g1s, zero4, zero4, 0);
#endif
    __builtin_amdgcn_s_wait_tensorcnt(0);
#else
    // -------- fallback: direct b128 gather (no TDM builtins available) ----
    (void)tile;
    const float4* __restrict__ t4 = (const float4*)table;
    float4* __restrict__ o4 = (float4*)out;
    constexpr int VPR = EMBED / 4;  // 192 float4 per row
    for (int r = 0; r < ROWS_PER_TDM; ++r) {
        const int id = ids[base + r];
        const float4* src = t4 + (long long)id * VPR;
        float4* dst = o4 + (long long)(base + r) * VPR;
        for (int col = (int)threadIdx.x; col < VPR; col += 32) {
            dst[col] = src[col];
        }
    }
#endif
}

extern "C" void kernel_launch(void* const* d_in, const int* in_sizes, int n_in,
                              void* d_out, int out_size, void* d_ws, size_t ws_size,
                              hipStream_t stream) {
    const int*   ids   = (const int*)d_in[0];     // [8192] int32
    const float* table = (const float*)d_in[1];   // [50257*768] fp32
    float*       out   = (float*)d_out;           // [8192*768] fp32

    const int num_tokens = in_sizes[0];           // 8 * 1024 = 8192
    const int groups = (num_tokens + ROWS_PER_TDM - 1) / ROWS_PER_TDM;  // 512

    embed_gather_kernel<<<groups, 32, 0, stream>>>(ids, table, out, num_tokens);
}